// EPAll2AllLayer_28595892257185
// MI455X (gfx1250) — compile-verified
//
#include <hip/hip_runtime.h>
#include <cstdint>

// EP all-to-all dispatch + combine (single-rank logical simulation).
// T=8192, TOPK=8, H=2048, E=64 per the reference; T/H derived from in_sizes.

#define EXPERTS 64
#define TOPK_K  8
#define CHUNK   256   // slots per chunk for the stable counting sort

typedef __attribute__((ext_vector_type(2))) float v2f;
typedef __attribute__((ext_vector_type(8))) float v8f;

// ---------------------------------------------------------------------------
// Kernel 1: per-(expert, chunk) histogram. Block c owns slots [c*256, c*256+256).
// Writes cnt[e*C + c] for all e (full overwrite -> no zero-init of ws needed).
// ---------------------------------------------------------------------------
__global__ __launch_bounds__(CHUNK) void ep_count_kernel(
    const int* __restrict__ eidx, int* __restrict__ cnt, int total, int C) {
  __shared__ int h[EXPERTS];
  const int c = blockIdx.x;
  if (threadIdx.x < EXPERTS) h[threadIdx.x] = 0;
  __syncthreads();
  const int i = c * CHUNK + threadIdx.x;
  if (i < total) atomicAdd(&h[eidx[i]], 1);   // integer LDS atomic: deterministic
  __syncthreads();
  if (threadIdx.x < EXPERTS) cnt[threadIdx.x * C + c] = h[threadIdx.x];
}

// ---------------------------------------------------------------------------
// Kernel 2: single-workgroup exclusive scan over cnt[E*C] (expert-major) plus
// per-expert totals (splits, written as float to d_out).
// ---------------------------------------------------------------------------
__global__ __launch_bounds__(256) void ep_scan_kernel(
    const int* __restrict__ cnt, int* __restrict__ scx,
    float* __restrict__ out_splits, int N, int C) {
  __shared__ int a[256];
  const int tid = threadIdx.x;
  const int seg = (N + 255) / 256;
  const int b = tid * seg;
  const int e = min(b + seg, N);
  int s = 0;
  for (int i = b; i < e; ++i) s += cnt[i];
  a[tid] = s;
  __syncthreads();
  // Hillis-Steele inclusive scan over 256 partials
  for (int off = 1; off < 256; off <<= 1) {
    int v = (tid >= off) ? a[tid - off] : 0;
    __syncthreads();
    a[tid] += v;
    __syncthreads();
  }
  int run = a[tid] - s;  // exclusive prefix of this thread's segment
  for (int i = b; i < e; ++i) { scx[i] = run; run += cnt[i]; }
  // splits[e] = total count for expert e
  if (tid < EXPERTS) {
    int t = 0;
    for (int c2 = 0; c2 < C; ++c2) t += cnt[tid * C + c2];
    out_splits[tid] = (float)t;   // counts <= 65536: exact in f32
  }
}

// ---------------------------------------------------------------------------
// Kernel 3: stable local rank within chunk -> final scatter row index.
// dst[i] = scanExcl[e_i*C + c] + #{j < i in chunk : e_j == e_i}
// ---------------------------------------------------------------------------
__global__ __launch_bounds__(CHUNK) void ep_rank_kernel(
    const int* __restrict__ eidx, const int* __restrict__ scx,
    int* __restrict__ dst, int total, int C) {
  __shared__ int se[CHUNK];
  const int c = blockIdx.x;
  const int i = c * CHUNK + threadIdx.x;
  const int e = (i < total) ? eidx[i] : -1;
  se[threadIdx.x] = e;
  __syncthreads();
  if (i < total) {
    int r = 0;
    for (int j = 0; j < (int)threadIdx.x; ++j) r += (se[j] == e);
    dst[i] = scx[e * C + c] + r;
  }
}

// ---------------------------------------------------------------------------
// Kernel 4: dispatch scatter via the CDNA5 async global<->LDS engine.
// One block per slot row (H=2048 f32 = 8 KB). Each lane moves its private 32 B:
// 2x global_load_async_to_lds_b128 -> s_wait_asynccnt 0 ->
// 2x global_store_async_from_lds_b128. The same inst offset applies to both the
// LDS and global address (ISA 15.18), so one LDS base VGPR covers both beats.
// Lane-private LDS slices -> no cross-thread dependency -> no barrier needed.
// ---------------------------------------------------------------------------
__global__ __launch_bounds__(256) void ep_dispatch_kernel(
    const float* __restrict__ x, const int* __restrict__ dst,
    float* __restrict__ disp, int H) {
  __shared__ float buf[2048];
  const int slot = blockIdx.x;
  const int t = slot / TOPK_K;
  const float* src = x + (size_t)t * H + (size_t)threadIdx.x * 8;
  float* d = disp + (size_t)dst[slot] * H + (size_t)threadIdx.x * 8;
  unsigned lds = (unsigned)(uintptr_t)(&buf[threadIdx.x * 8]);
  asm volatile(
      "global_load_async_to_lds_b128 %0, %1, off\n\t"
      "global_load_async_to_lds_b128 %0, %1, off offset:16\n\t"
      "s_wait_asynccnt 0\n\t"
      "global_store_async_from_lds_b128 %2, %0, off\n\t"
      "global_store_async_from_lds_b128 %2, %0, off offset:16\n\t"
      "s_wait_asynccnt 0"
      :
      : "v"(lds), "v"(src), "v"(d)
      : "memory");
}

// ---------------------------------------------------------------------------
// Kernel 5: routing-weight row sums via WMMA. The combine einsum factorizes as
// combined[t,h] = (sum_k w[t,k]) * x[t,h]; the k-contraction for a 16-token
// tile is W[16x8] x ones[8x16] = two chained V_WMMA_F32_16X16X4_F32 ops
// (k=0..3 with C=0, then k=4..7 accumulating D->C). One full wave32 per tile
// (EXEC all-ones per WMMA restrictions).
// A layout (32-bit 16x4): lanes 0-15: VGPR0=K0,VGPR1=K1 ; lanes 16-31: K2,K3.
// D layout (32-bit 16x16): VGPR j = row M=j (lanes 0-15) / M=8+j (lanes 16-31).
// ---------------------------------------------------------------------------
__global__ __launch_bounds__(32) void ep_wsum_kernel(
    const float* __restrict__ w, float* __restrict__ sw, int T) {
  const int lane = threadIdx.x;          // 0..31, one wave per block
  const int t0 = blockIdx.x * 16;
  const int row = t0 + (lane & 15);
  const int hi = lane >> 4;              // 0: K=0..1 / 1: K=2..3
  const float* wr = w + (size_t)row * TOPK_K;
  v2f a0, a1, b;
  a0.x = wr[hi * 2 + 0];
  a0.y = wr[hi * 2 + 1];
  a1.x = wr[4 + hi * 2 + 0];
  a1.y = wr[4 + hi * 2 + 1];
  b.x = 1.0f;                            // B = ones: layout-invariant
  b.y = 1.0f;
  v8f c = {};
  c = __builtin_amdgcn_wmma_f32_16x16x4_f32(false, a0, false, b, (short)0, c,
                                            false, false);
  c = __builtin_amdgcn_wmma_f32_16x16x4_f32(false, a1, false, b, (short)0, c,
                                            false, false);
  // every column N holds the same row sums; lane N=0 of each half writes them
  if ((lane & 15) == 0) {
#pragma unroll
    for (int j = 0; j < 8; ++j) sw[t0 + hi * 8 + j] = c[j];
  }
}

// ---------------------------------------------------------------------------
// Kernel 6: combine. gathered[t,k,:] == x[t,:] (round-trip permutation), so
// combined[t,h] = sw[t] * x[t,h] with sw from the WMMA row-sum kernel.
// ---------------------------------------------------------------------------
__global__ __launch_bounds__(256) void ep_combine_kernel(
    const float* __restrict__ x, const float* __restrict__ sw,
    float* __restrict__ outc, int H) {
  const int t = blockIdx.x;
  const float s = sw[t];
  const float* xr = x + (size_t)t * H + (size_t)threadIdx.x * 8;
  float* orow = outc + (size_t)t * H + (size_t)threadIdx.x * 8;
  float4 v0 = *(const float4*)(xr);
  float4 v1 = *(const float4*)(xr + 4);
  *(float4*)(orow) = make_float4(s * v0.x, s * v0.y, s * v0.z, s * v0.w);
  *(float4*)(orow + 4) = make_float4(s * v1.x, s * v1.y, s * v1.z, s * v1.w);
}

// ---------------------------------------------------------------------------
extern "C" void kernel_launch(void* const* d_in, const int* in_sizes, int n_in,
                              void* d_out, int out_size, void* d_ws, size_t ws_size,
                              hipStream_t stream) {
  const float* x    = (const float*)d_in[0];   // [T, H] f32
  const float* w    = (const float*)d_in[1];   // [T, K] f32
  const int*   eidx = (const int*)d_in[2];     // [T, K] i32

  const int total = in_sizes[1];               // T*K
  const int T = total / TOPK_K;
  const int H = in_sizes[0] / T;
  const int C = (total + CHUNK - 1) / CHUNK;   // #chunks
  const int N = EXPERTS * C;                   // count-matrix size

  // workspace: cnt[E*C] | scanExcl[E*C] | dst[T*K] | sw[T]  (~0.45 MB)
  int* cnt = (int*)d_ws;
  int* scx = cnt + N;
  int* dst = scx + N;
  float* sw = (float*)(dst + total);

  // outputs, concatenated flat in return order (all as f32)
  float* disp   = (float*)d_out;                       // [T*K, H]
  float* splits = disp + (size_t)total * (size_t)H;    // [E]
  float* comb   = splits + EXPERTS;                    // [T, H]

  ep_count_kernel<<<C, CHUNK, 0, stream>>>(eidx, cnt, total, C);
  ep_scan_kernel<<<1, 256, 0, stream>>>(cnt, scx, splits, N, C);
  ep_rank_kernel<<<C, CHUNK, 0, stream>>>(eidx, scx, dst, total, C);
  ep_dispatch_kernel<<<total, 256, 0, stream>>>(x, dst, disp, H);
  ep_wsum_kernel<<<(T + 15) / 16, 32, 0, stream>>>(w, sw, T);
  ep_combine_kernel<<<T, 256, 0, stream>>>(x, sw, comb, H);
}